// MLFF_78838419685604
// MI455X (gfx1250) — compile-verified
//
#include <hip/hip_runtime.h>

typedef __attribute__((ext_vector_type(2))) float v2f;
typedef __attribute__((ext_vector_type(8))) float v8f;
typedef __attribute__((ext_vector_type(4))) unsigned int v4u;
typedef __attribute__((ext_vector_type(8))) int v8i;
typedef __attribute__((ext_vector_type(4))) int v4i;

#define NFEAT 42
#define D0 256
#define D1 128
#define D2 64
#define NBATCH 8
#define NA 512
#define NATOMS (NBATCH * NA)   // 4096
#define MNBR 100
#define SLICE (MNBR * NFEAT * 3)   // 12600 floats / (b,i)
#define WPB 2                  // waves per block in mlp kernel

// padded LDS row strides (floats) to avoid 64-bank conflicts
#define S0 (D0 + 2)            // 258
#define S1 (D1 + 2)            // 130
#define S2 (D2 + 2)            // 66
#define LDSW (16 * (S0 + S1 + S2))   // floats per wave

__device__ __forceinline__ v8f wmma4(v2f a, v2f b, v8f c) {
  // V_WMMA_F32_16X16X4_F32 : D = A(16x4) * B(4x16) + C(16x16), wave32
  return __builtin_amdgcn_wmma_f32_16x16x4_f32(
      /*neg_a=*/false, a, /*neg_b=*/false, b,
      /*c_mod=*/(short)0, c, /*reuse_a=*/false, /*reuse_b=*/false);
}

// fast sigmoid: v_exp_f32 + v_rcp_f32 (avoid IEEE div_scale/div_fmas chains)
__device__ __forceinline__ float sigm(float x) {
  return __builtin_amdgcn_rcpf(1.0f + __expf(-x));
}

// One wave == one 16-atom tile. Full fwd + bwd through the MLP with f32 WMMA.
__global__ __launch_bounds__(32 * WPB) void mlp_kernel(
    const float* __restrict__ image,
    const float* __restrict__ W0, const float* __restrict__ b0,
    const float* __restrict__ W1, const float* __restrict__ b1,
    const float* __restrict__ W2, const float* __restrict__ b2,
    const float* __restrict__ W3, const float* __restrict__ b3,
    float* __restrict__ Ei_out,        // [4096]
    float* __restrict__ dE_ws)         // [4096 * 42]
{
  __shared__ float lds[WPB * LDSW];
  const int wave = threadIdx.x >> 5;
  const int lane = threadIdx.x & 31;
  const int half = lane >> 4;          // 0: lanes 0-15, 1: lanes 16-31
  const int lm   = lane & 15;

  float* H0 = &lds[wave * LDSW];       // 16 x S0 (activations layer0, later grad)
  float* H1 = H0 + 16 * S0;            // 16 x S1
  float* H2 = H1 + 16 * S1;            // 16 x S2

  const int tile  = blockIdx.x * WPB + wave;
  const int atom0 = tile * 16;

  // ---------------- Layer 0: X[16x42] @ W0[42x256], sigmoid -> H0 ------------
  {
    const float* Xrow = image + (size_t)(atom0 + lm) * NFEAT;
    v2f afrag[11];                      // A fragments reused across all n-tiles
#pragma unroll
    for (int t = 0; t < 11; ++t) {
      const int k0 = 4 * t + 2 * half;
      afrag[t].x = (k0 + 0 < NFEAT) ? Xrow[k0 + 0] : 0.0f;
      afrag[t].y = (k0 + 1 < NFEAT) ? Xrow[k0 + 1] : 0.0f;
    }
    for (int nb = 0; nb < D0; nb += 16) {
      v8f acc = {};
#pragma unroll
      for (int t = 0; t < 11; ++t) {
        const int k0 = 4 * t + 2 * half;
        v2f b;
        b.x = (k0 + 0 < NFEAT) ? W0[(k0 + 0) * D0 + nb + lm] : 0.0f;
        b.y = (k0 + 1 < NFEAT) ? W0[(k0 + 1) * D0 + nb + lm] : 0.0f;
        acc = wmma4(afrag[t], b, acc);
      }
      const float bias = b0[nb + lm];
#pragma unroll
      for (int r = 0; r < 8; ++r) {
        const int m = r + 8 * half;
        H0[m * S0 + nb + lm] = sigm(acc[r] + bias);
      }
    }
  }
  __syncthreads();

  // ---------------- Layer 1: H0[16x256] @ W1[256x128], sigmoid -> H1 ---------
  for (int nb = 0; nb < D1; nb += 16) {
    v8f acc = {};
#pragma unroll 4
    for (int kb = 0; kb < D0; kb += 4) {
      const int k0 = kb + 2 * half;
      v2f a, b;
      a.x = H0[lm * S0 + k0 + 0];
      a.y = H0[lm * S0 + k0 + 1];
      b.x = W1[(k0 + 0) * D1 + nb + lm];
      b.y = W1[(k0 + 1) * D1 + nb + lm];
      acc = wmma4(a, b, acc);
    }
    const float bias = b1[nb + lm];
#pragma unroll
    for (int r = 0; r < 8; ++r) {
      const int m = r + 8 * half;
      H1[m * S1 + nb + lm] = sigm(acc[r] + bias);
    }
  }
  __syncthreads();

  // ---------------- Layer 2: H1[16x128] @ W2[128x64], sigmoid -> H2 ----------
  for (int nb = 0; nb < D2; nb += 16) {
    v8f acc = {};
#pragma unroll 4
    for (int kb = 0; kb < D1; kb += 4) {
      const int k0 = kb + 2 * half;
      v2f a, b;
      a.x = H1[lm * S1 + k0 + 0];
      a.y = H1[lm * S1 + k0 + 1];
      b.x = W2[(k0 + 0) * D2 + nb + lm];
      b.y = W2[(k0 + 1) * D2 + nb + lm];
      acc = wmma4(a, b, acc);
    }
    const float bias = b2[nb + lm];
#pragma unroll
    for (int r = 0; r < 8; ++r) {
      const int m = r + 8 * half;
      H2[m * S2 + nb + lm] = sigm(acc[r] + bias);
    }
  }
  __syncthreads();

  // ---------------- Output energy: Ei = H2 @ W3 + b3 -------------------------
  if (half == 0) {
    float e = b3[0];
    for (int k = 0; k < D2; ++k) e += H2[lm * S2 + k] * W3[k];
    Ei_out[atom0 + lm] = e;
  }
  __syncthreads();

  // ================= Backward ================================================
  // g_z2 = W3 (*) h2*(1-h2)    (overwrite H2)
  for (int i = lane; i < 16 * D2; i += 32) {
    const int m = i / D2, k = i - m * D2;
    const float h = H2[m * S2 + k];
    H2[m * S2 + k] = W3[k] * h * (1.0f - h);
  }
  __syncthreads();

  // g_z1 = (g_z2 @ W2^T) (*) h1*(1-h1)    B'[k][n] = W2[n*D2 + k]  (overwrite H1)
  for (int nb = 0; nb < D1; nb += 16) {
    v8f acc = {};
#pragma unroll 4
    for (int kb = 0; kb < D2; kb += 4) {
      const int k0 = kb + 2 * half;
      v2f a, b;
      a.x = H2[lm * S2 + k0 + 0];
      a.y = H2[lm * S2 + k0 + 1];
      b.x = W2[(nb + lm) * D2 + k0 + 0];
      b.y = W2[(nb + lm) * D2 + k0 + 1];
      acc = wmma4(a, b, acc);
    }
#pragma unroll
    for (int r = 0; r < 8; ++r) {
      const int m = r + 8 * half, n = nb + lm;
      const float h = H1[m * S1 + n];
      H1[m * S1 + n] = acc[r] * h * (1.0f - h);
    }
  }
  __syncthreads();

  // g_z0 = (g_z1 @ W1^T) (*) h0*(1-h0)    B'[k][n] = W1[n*D1 + k]  (overwrite H0)
  for (int nb = 0; nb < D0; nb += 16) {
    v8f acc = {};
#pragma unroll 4
    for (int kb = 0; kb < D1; kb += 4) {
      const int k0 = kb + 2 * half;
      v2f a, b;
      a.x = H1[lm * S1 + k0 + 0];
      a.y = H1[lm * S1 + k0 + 1];
      b.x = W1[(nb + lm) * D1 + k0 + 0];
      b.y = W1[(nb + lm) * D1 + k0 + 1];
      acc = wmma4(a, b, acc);
    }
#pragma unroll
    for (int r = 0; r < 8; ++r) {
      const int m = r + 8 * half, n = nb + lm;
      const float h = H0[m * S0 + n];
      H0[m * S0 + n] = acc[r] * h * (1.0f - h);
    }
  }
  __syncthreads();

  // dE/dx[16x42] = g_z0 @ W0^T    B'[k][n] = W0[n*D0 + k], n padded 42 -> 48
  for (int nb = 0; nb < 48; nb += 16) {
    const int n = nb + lm;
    const bool nval = (n < NFEAT);
    v8f acc = {};
#pragma unroll 4
    for (int kb = 0; kb < D0; kb += 4) {
      const int k0 = kb + 2 * half;
      v2f a, b;
      a.x = H0[lm * S0 + k0 + 0];
      a.y = H0[lm * S0 + k0 + 1];
      b.x = nval ? W0[n * D0 + k0 + 0] : 0.0f;
      b.y = nval ? W0[n * D0 + k0 + 1] : 0.0f;
      acc = wmma4(a, b, acc);
    }
    if (nval) {
#pragma unroll
      for (int r = 0; r < 8; ++r) {
        const int m = r + 8 * half;
        dE_ws[(size_t)(atom0 + m) * NFEAT + n] = acc[r];
      }
    }
  }
}

// Per-batch energy sum: Etot[b] = sum_n Ei[b,n]
__global__ __launch_bounds__(256) void etot_kernel(const float* __restrict__ Ei,
                                                   float* __restrict__ Etot) {
  __shared__ float red[256];
  const int b = blockIdx.x, tid = threadIdx.x;
  float s = 0.0f;
  for (int n = tid; n < NA; n += 256) s += Ei[b * NA + n];
  red[tid] = s;
  __syncthreads();
  for (int off = 128; off > 0; off >>= 1) {
    if (tid < off) red[tid] += red[tid + off];
    __syncthreads();
  }
  if (tid == 0) Etot[b] = red[0];
}

// Force[b,i,d] = sum_m mask * sum_f dE[b, nbr-1, f] * dfeat[b,i,m,f,d]
// One block per (b,i). The Tensor Data Mover DMAs the contiguous 50.4 KB
// dfeat slice into LDS (one TDM op instead of ~3150 vector-load issues);
// threads then contract out of LDS while dE rows come from L2.
__global__ __launch_bounds__(256) void force_kernel(
    const float* __restrict__ dfeat, const int* __restrict__ neighbor,
    const float* __restrict__ dE, float* __restrict__ Force)
{
  __shared__ float dfs[SLICE];          // 50400 B
  __shared__ int   nbr[MNBR];
  __shared__ float red[3 * 256];
  const int bi  = blockIdx.x;           // b*NA + i
  const int b   = bi / NA;
  const int tid = threadIdx.x;

  if (tid < MNBR) nbr[tid] = neighbor[(size_t)bi * MNBR + tid];

  // ---- TDM: tensor_load_to_lds of one 1x12600 f32 tile --------------------
  if (tid < 32) {                       // one wave issues the DMA (EXEC ignored)
    const unsigned long long ga =
        (unsigned long long)(const void*)(dfeat + (size_t)bi * SLICE);
    const unsigned lds_addr = (unsigned)(size_t)(void*)dfs;  // LDS byte offset
    v4u g0;                             // D# group 0 (ISA 8.3)
    g0.x = 1u;                          // count=1, user descriptor
    g0.y = lds_addr;                    // lds_addr [63:32]
    g0.z = (unsigned)(ga & 0xFFFFFFFFull);        // global_addr [95:64]
    g0.w = (unsigned)(ga >> 32) | (2u << 30);     // addr[56:32] | type=2
    v8i g1;                             // D# group 1 (ISA 8.4)
    g1[0] = 2 << 16;                    // data_size=4B, wg_mask=0
    g1[1] = (SLICE & 0xFFFF) << 16;     // tensor_dim0[15:0]
    g1[2] = (SLICE >> 16) | (1 << 16);  // tensor_dim0[31:16] | tensor_dim1=1
    g1[3] = (SLICE & 0xFFFF) << 16;     // tile_dim0 = 12600
    g1[4] = 1;                          // tile_dim1=1, tile_dim2=0
    g1[5] = SLICE;                      // tensor_dim0_stride[31:0]
    g1[6] = 0; g1[7] = 0;
    v4i g2 = {0, 0, 0, 0};              // <=2D: groups 2/3 unused
    v4i g3 = {0, 0, 0, 0};
#if __clang_major__ >= 23
    v8i gx = {0, 0, 0, 0, 0, 0, 0, 0};
    __builtin_amdgcn_tensor_load_to_lds(g0, g1, g2, g3, gx, 0);
#else
    __builtin_amdgcn_tensor_load_to_lds(g0, g1, g2, g3, 0);
#endif
  }
  __builtin_amdgcn_s_wait_tensorcnt(0);
  __syncthreads();

  float a0 = 0.0f, a1 = 0.0f, a2 = 0.0f;
  for (int w = tid; w < SLICE; w += 256) {
    const int m = w / (NFEAT * 3);
    const int j = w - m * (NFEAT * 3);
    const int f = j / 3;
    const int d = j - f * 3;
    const int nb = nbr[m];
    if (nb > 0) {
      const float p = dfs[w] * dE[((size_t)b * NA + (nb - 1)) * NFEAT + f];
      if (d == 0) a0 += p; else if (d == 1) a1 += p; else a2 += p;
    }
  }
  red[tid] = a0; red[256 + tid] = a1; red[512 + tid] = a2;
  __syncthreads();
  for (int off = 128; off > 0; off >>= 1) {
    if (tid < off) {
      red[tid]       += red[tid + off];
      red[256 + tid] += red[256 + tid + off];
      red[512 + tid] += red[512 + tid + off];
    }
    __syncthreads();
  }
  if (tid < 3) Force[(size_t)bi * 3 + tid] = red[tid * 256];
}

extern "C" void kernel_launch(void* const* d_in, const int* in_sizes, int n_in,
                              void* d_out, int out_size, void* d_ws, size_t ws_size,
                              hipStream_t stream) {
  const float* image    = (const float*)d_in[0];
  const float* dfeat    = (const float*)d_in[1];
  const int*   neighbor = (const int*)  d_in[2];
  const float* W0 = (const float*)d_in[3];  const float* b0 = (const float*)d_in[4];
  const float* W1 = (const float*)d_in[5];  const float* b1 = (const float*)d_in[6];
  const float* W2 = (const float*)d_in[7];  const float* b2 = (const float*)d_in[8];
  const float* W3 = (const float*)d_in[9];  const float* b3 = (const float*)d_in[10];

  float* out   = (float*)d_out;
  float* Etot  = out;                    // [8]
  float* Ei    = out + NBATCH;           // [4096]
  float* Force = out + NBATCH + NATOMS;  // [4096*3]
  float* dE_ws = (float*)d_ws;           // [4096*42] floats (688 KB)

  mlp_kernel<<<NATOMS / (16 * WPB), 32 * WPB, 0, stream>>>(
      image, W0, b0, W1, b1, W2, b2, W3, b3, Ei, dE_ws);
  etot_kernel<<<NBATCH, 256, 0, stream>>>(Ei, Etot);
  force_kernel<<<NATOMS, 256, 0, stream>>>(dfeat, neighbor, dE_ws, Force);
}